// LAMMPS_MACE_34832184770823
// MI455X (gfx1250) — compile-verified
//
#include <hip/hip_runtime.h>
#include <hip/hip_bf16.h>

typedef float v2f __attribute__((ext_vector_type(2)));
typedef float v8f __attribute__((ext_vector_type(8)));

#define N_ATOMS 100000
#define N_EDGES 3200000
#define HIDDEN 16

// output layout (floats): [0]=TE, [1..100001) node_e, [100001..400001) forces,
// [400001..400010) virials, [400010..1300010) atom_virial
#define OUT_NE   1
#define OUT_F    100001
#define OUT_VIR  400001
#define OUT_AV   400010

__device__ __forceinline__ float red16(float v) {
    // sum across each 16-lane group of a wave32
    v += __shfl_xor(v, 1, 32);
    v += __shfl_xor(v, 2, 32);
    v += __shfl_xor(v, 4, 32);
    v += __shfl_xor(v, 8, 32);
    return v;
}

// fast sigmoid: v_exp_f32 + v_add + v_rcp_f32 (no IEEE div sequence)
__device__ __forceinline__ float sigmoidf(float x) {
    return __builtin_amdgcn_rcpf(1.0f + __expf(-x));
}

// ---------------- K1: zero accumulators ----------------
__global__ void k_zero(float* __restrict__ h, float* __restrict__ out) {
    int idx = blockIdx.x * blockDim.x + threadIdx.x;
    const int HTOT = N_ATOMS * HIDDEN;          // 1,600,000
    const int OTOT = 900010;                     // TE + virials + atom_virial
    if (idx < HTOT) {
        h[idx] = 0.0f;
    } else if (idx < HTOT + OTOT) {
        int k = idx - HTOT;
        if (k == 0) out[0] = 0.0f;
        else        out[400000 + k] = 0.0f;      // covers 400001..1300009
    }
}

// ---------------- K2: edge forward: msg = silu(phi@W1+b1); h[recv] += msg ----
__global__ void k_edge_fwd(const float* __restrict__ vectors,
                           const int*   __restrict__ receiver,
                           const float* __restrict__ W1,
                           const float* __restrict__ b1p,
                           const float* __restrict__ alphas,
                           float* __restrict__ h) {
    const int lane = threadIdx.x & 31;
    const int wave = threadIdx.x >> 5;
    const int m  = lane & 15;
    const int hi = lane >> 4;
    const int base = (blockIdx.x * 8 + wave) * 16;   // 16 edges per wave

    // r2 of edge base+m (computed redundantly by the two half-waves)
    const int e0 = base + m;
    float vx = vectors[3 * e0 + 0], vy = vectors[3 * e0 + 1], vz = vectors[3 * e0 + 2];
    float r2 = vx * vx + vy * vy + vz * vz;

    // A (16x4 f32 layout): VGPR0 holds K=2*hi, VGPR1 holds K=2*hi+1
    v2f a1, a2, bA, bB;
    a1.x = __expf(-r2 * alphas[2 * hi + 0]);
    a1.y = __expf(-r2 * alphas[2 * hi + 1]);
    a2.x = __expf(-r2 * alphas[2 * hi + 4]);
    a2.y = __expf(-r2 * alphas[2 * hi + 5]);
    // B (4x16): VGPR0 row K=2*hi, VGPR1 row K=2*hi+1, col = m
    bA.x = W1[(2 * hi + 0) * 16 + m];
    bA.y = W1[(2 * hi + 1) * 16 + m];
    bB.x = W1[(2 * hi + 4) * 16 + m];
    bB.y = W1[(2 * hi + 5) * 16 + m];

    v8f c = {};
    c = __builtin_amdgcn_wmma_f32_16x16x4_f32(false, a1, false, bA, (short)0, c, false, false);
    c = __builtin_amdgcn_wmma_f32_16x16x4_f32(false, a2, false, bB, (short)0, c, false, false);

    const float bias = b1p[m];
#pragma unroll
    for (int j = 0; j < 8; ++j) {
        float x = c[j] + bias;                 // pre-activation, edge row j+8*hi, col m
        float msg = x * sigmoidf(x);           // silu
        int e = base + j + 8 * hi;
        int r = receiver[e];
        atomicAdd(&h[r * HIDDEN + m], msg);
    }
}

// ---------------- K3: node MLP + energy + forces + g ----------------
__global__ void k_node(const float* __restrict__ h,
                       const float* __restrict__ positions,
                       const float* __restrict__ localg,
                       const float* __restrict__ W2,
                       const float* __restrict__ w3,
                       const float* __restrict__ field,
                       float* __restrict__ out,
                       float* __restrict__ g) {
    __shared__ float tS[8][16][17];
    __shared__ float teS;

    const int lane = threadIdx.x & 31;
    const int wave = threadIdx.x >> 5;
    const int m  = lane & 15;
    const int hi = lane >> 4;
    const int base = (blockIdx.x * 8 + wave) * 16;   // 16 atoms per wave

    if (threadIdx.x == 0) teS = 0.0f;
    __syncthreads();

    const int am = min(base + m, N_ATOMS - 1);       // clamped A-row load

    // u = h @ W2  (M=16 atoms, K=16, N=16) via 4 f32 WMMAs
    v8f c = {};
#pragma unroll
    for (int t = 0; t < 4; ++t) {
        v2f a, b;
        a.x = h[am * HIDDEN + 4 * t + 2 * hi + 0];
        a.y = h[am * HIDDEN + 4 * t + 2 * hi + 1];
        b.x = W2[(4 * t + 2 * hi + 0) * 16 + m];
        b.y = W2[(4 * t + 2 * hi + 1) * 16 + m];
        c = __builtin_amdgcn_wmma_f32_16x16x4_f32(false, a, false, b, (short)0, c, false, false);
    }

    const float fx = field[0], fy = field[1], fz = field[2];
    const float w3m = w3[m];

#pragma unroll
    for (int j = 0; j < 8; ++j) {
        float x = c[j];                        // u[row=j+8*hi][m]
        float s = sigmoidf(x);
        float sil = x * s;                     // silu
        float dsil = s * (1.0f + x * (1.0f - s));
        tS[wave][j + 8 * hi][m] = dsil * w3m;  // t = silu'(u) * w3
        float er = red16(sil * w3m);           // node energy (pre position term)
        if (m == j) {
            int a = base + j + 8 * hi;
            if (a < N_ATOMS) {
                float pe = positions[3 * a] * fx + positions[3 * a + 1] * fy + positions[3 * a + 2] * fz;
                float ne = er + pe;
                float loc = localg[a];
                out[OUT_NE + a] = ne;
                atomicAdd(&teS, loc * ne);     // block-local energy accumulation (LDS)
                out[OUT_F + 3 * a + 0] = -loc * fx;
                out[OUT_F + 3 * a + 1] = -loc * fy;
                out[OUT_F + 3 * a + 2] = -loc * fz;
            }
        }
    }
    __syncthreads();

    if (threadIdx.x == 0) atomicAdd(&out[0], teS);   // one global atomic per block

    // g = local * (t @ W2^T): A = t (from LDS transpose), B[k][n] = W2[n][k]
    v8f gc = {};
#pragma unroll
    for (int t = 0; t < 4; ++t) {
        v2f a, b;
        a.x = tS[wave][m][4 * t + 2 * hi + 0];
        a.y = tS[wave][m][4 * t + 2 * hi + 1];
        b.x = W2[m * 16 + 4 * t + 2 * hi + 0];
        b.y = W2[m * 16 + 4 * t + 2 * hi + 1];
        gc = __builtin_amdgcn_wmma_f32_16x16x4_f32(false, a, false, b, (short)0, gc, false, false);
    }
#pragma unroll
    for (int j = 0; j < 8; ++j) {
        int a = base + j + 8 * hi;
        if (a < N_ATOMS) {
            g[a * HIDDEN + m] = gc[j] * localg[a];
        }
    }
}

// ---------------- K4: edge backward: s_e, virial scatter ----------------
__global__ void k_edge_bwd(const float* __restrict__ vectors,
                           const int*   __restrict__ sender,
                           const int*   __restrict__ receiver,
                           const float* __restrict__ W1,
                           const float* __restrict__ b1p,
                           const float* __restrict__ alphas,
                           const float* __restrict__ g,
                           float* __restrict__ out) {
    __shared__ float dS[8][16][17];
    __shared__ float r2S[8][16];
    __shared__ float sS[8][16];

    const int lane = threadIdx.x & 31;
    const int wave = threadIdx.x >> 5;
    const int m  = lane & 15;
    const int hi = lane >> 4;
    const int base = (blockIdx.x * 8 + wave) * 16;

    // recompute pre = phi@W1 + b1 (same 2 WMMAs as forward)
    const int e0 = base + m;
    float vx = vectors[3 * e0 + 0], vy = vectors[3 * e0 + 1], vz = vectors[3 * e0 + 2];
    float r2 = vx * vx + vy * vy + vz * vz;
    if (hi == 0) r2S[wave][m] = r2;

    v2f a1, a2, bA, bB;
    a1.x = __expf(-r2 * alphas[2 * hi + 0]);
    a1.y = __expf(-r2 * alphas[2 * hi + 1]);
    a2.x = __expf(-r2 * alphas[2 * hi + 4]);
    a2.y = __expf(-r2 * alphas[2 * hi + 5]);
    bA.x = W1[(2 * hi + 0) * 16 + m];
    bA.y = W1[(2 * hi + 1) * 16 + m];
    bB.x = W1[(2 * hi + 4) * 16 + m];
    bB.y = W1[(2 * hi + 5) * 16 + m];

    v8f c = {};
    c = __builtin_amdgcn_wmma_f32_16x16x4_f32(false, a1, false, bA, (short)0, c, false, false);
    c = __builtin_amdgcn_wmma_f32_16x16x4_f32(false, a2, false, bB, (short)0, c, false, false);

    const float bias = b1p[m];
#pragma unroll
    for (int j = 0; j < 8; ++j) {
        float x = c[j] + bias;                 // pre[row=j+8hi][m]
        int e = base + j + 8 * hi;
        float gv = g[receiver[e] * HIDDEN + m];
        float s = sigmoidf(x);
        float dsil = s * (1.0f + x * (1.0f - s));
        dS[wave][j + 8 * hi][m] = gv * dsil;   // dpre
    }
    __syncthreads();

    // dphi = dpre @ W1^T  (B cols n>=8 zero-padded => dphi cols n>=8 are 0)
    v8f dphi = {};
#pragma unroll
    for (int t = 0; t < 4; ++t) {
        v2f a, b;
        a.x = dS[wave][m][4 * t + 2 * hi + 0];
        a.y = dS[wave][m][4 * t + 2 * hi + 1];
        b.x = (m < 8) ? W1[m * 16 + 4 * t + 2 * hi + 0] : 0.0f;
        b.y = (m < 8) ? W1[m * 16 + 4 * t + 2 * hi + 1] : 0.0f;
        dphi = __builtin_amdgcn_wmma_f32_16x16x4_f32(false, a, false, b, (short)0, dphi, false, false);
    }

    const float af = alphas[m & 7];
#pragma unroll
    for (int j = 0; j < 8; ++j) {
        float r2r = r2S[wave][j + 8 * hi];
        float term = dphi[j] * (-af * __expf(-r2r * af));  // dphi==0 for m>=8
        float sr = red16(term);                             // s_e for edge row j+8hi
        if (m == j) sS[wave][j + 8 * hi] = sr;
    }
    __syncthreads();

    // per-edge virial scatter: atom_virial[snd] += s*v v^T ; same for rcv
    if (lane < 16) {
        int e = base + lane;
        float s = sS[wave][lane];
        float v[3];
        v[0] = vectors[3 * e + 0];
        v[1] = vectors[3 * e + 1];
        v[2] = vectors[3 * e + 2];
        int snd = sender[e];
        int rcv = receiver[e];
        float* avs = &out[OUT_AV + snd * 9];
        float* avr = &out[OUT_AV + rcv * 9];
#pragma unroll
        for (int i = 0; i < 3; ++i) {
#pragma unroll
            for (int jj = 0; jj < 3; ++jj) {
                float val = s * v[i] * v[jj];
                atomicAdd(&avs[3 * i + jj], val);
                atomicAdd(&avr[3 * i + jj], val);
            }
        }
    }
}

// ---------------- K5: virials = sum over atoms of atom_virial ----------------
__global__ void k_virial_reduce(const float* __restrict__ av, float* __restrict__ vir) {
    __shared__ float red[9];
    if (threadIdx.x < 9) red[threadIdx.x] = 0.0f;
    __syncthreads();
    float acc[9] = {0, 0, 0, 0, 0, 0, 0, 0, 0};
    for (int a = blockIdx.x * blockDim.x + threadIdx.x; a < N_ATOMS; a += gridDim.x * blockDim.x) {
#pragma unroll
        for (int cc = 0; cc < 9; ++cc) acc[cc] += av[a * 9 + cc];
    }
#pragma unroll
    for (int cc = 0; cc < 9; ++cc) atomicAdd(&red[cc], acc[cc]);
    __syncthreads();
    if (threadIdx.x < 9) atomicAdd(&vir[threadIdx.x], red[threadIdx.x]);
}

extern "C" void kernel_launch(void* const* d_in, const int* in_sizes, int n_in,
                              void* d_out, int out_size, void* d_ws, size_t ws_size,
                              hipStream_t stream) {
    const float* positions = (const float*)d_in[0];
    const float* vectors   = (const float*)d_in[1];
    const float* localg    = (const float*)d_in[2];
    const float* W1        = (const float*)d_in[3];
    const float* b1p       = (const float*)d_in[4];
    const float* W2        = (const float*)d_in[5];
    const float* w3        = (const float*)d_in[6];
    const float* field     = (const float*)d_in[7];
    const float* alphas    = (const float*)d_in[8];
    const int*   edge_idx  = (const int*)d_in[9];
    float* out = (float*)d_out;

    float* h = (float*)d_ws;                 // N_ATOMS*16 floats
    float* g = h + (size_t)N_ATOMS * HIDDEN; // N_ATOMS*16 floats

    const int* sender   = edge_idx;
    const int* receiver = edge_idx + N_EDGES;

    // K1: zero h + accumulated out regions (TE, virials, atom_virial)
    k_zero<<<9766, 256, 0, stream>>>(h, out);
    // K2: edge forward (16 edges / wave), E/128 = 25000 blocks exactly
    k_edge_fwd<<<25000, 256, 0, stream>>>(vectors, receiver, W1, b1p, alphas, h);
    // K3: node MLP (16 atoms / wave), ceil(100000/128) = 782 blocks
    k_node<<<782, 256, 0, stream>>>(h, positions, localg, W2, w3, field, out, g);
    // K4: edge backward
    k_edge_bwd<<<25000, 256, 0, stream>>>(vectors, sender, receiver, W1, b1p, alphas, g, out);
    // K5: reduce atom_virial -> virials
    k_virial_reduce<<<120, 256, 0, stream>>>(out + OUT_AV, out + OUT_VIR);
}